// DTWMatcher_22600117911558
// MI455X (gfx1250) — compile-verified
//
#include <hip/hip_runtime.h>
#include <math.h>

// ---------------------------------------------------------------------------
// DTW matcher forward for MI455X (gfx1250, wave32).
// Pipeline: zero out -> invnorms -> fp32 WMMA GEMM (sim) -> radix-select dropline ->
//           NW forward DP (LDS cummin scan) -> traceback.
// ---------------------------------------------------------------------------

#define Nx 4096
#define Mx 4096
#define Dd 1024
#define TOTAL 16777216u          // N*M
#define TARGET_IDX 11744052u     // TOTAL - k, k = int(TOTAL*0.3) = 5033164

typedef __attribute__((ext_vector_type(2))) float v2f;
typedef __attribute__((ext_vector_type(8))) float v8f;

__device__ __forceinline__ unsigned f2key(float f) {
    unsigned u = __float_as_uint(f);
    return u ^ ((u & 0x80000000u) ? 0xFFFFFFFFu : 0x80000000u);
}
__device__ __forceinline__ float key2f(unsigned k) {
    unsigned u = (k & 0x80000000u) ? (k ^ 0x80000000u) : (k ^ 0xFFFFFFFFu);
    return __uint_as_float(u);
}

// ---------------- zero helpers (graph replays leave stale state) -----------
__global__ void kzero_f(float* __restrict__ p, unsigned n) {
    unsigned i = blockIdx.x * blockDim.x + threadIdx.x;
    unsigned stride = gridDim.x * blockDim.x;
    for (; i < n; i += stride) p[i] = 0.0f;
}
__global__ void kzero_u(unsigned* __restrict__ p, unsigned n) {
    unsigned i = blockIdx.x * blockDim.x + threadIdx.x;
    unsigned stride = gridDim.x * blockDim.x;
    for (; i < n; i += stride) p[i] = 0u;
}

// ---------------- inverse row norms ----------------------------------------
__global__ __launch_bounds__(256) void knorm(const float* __restrict__ z,
                                             const float* __restrict__ x,
                                             float* __restrict__ invz,
                                             float* __restrict__ invx) {
    int row = blockIdx.x;  // 0..8191 : first 4096 -> z, rest -> x
    const float* src = (row < Nx) ? (z + (size_t)row * Dd)
                                  : (x + (size_t)(row - Nx) * Dd);
    float s = 0.0f;
    for (int i = threadIdx.x; i < Dd; i += 256) {
        float v = src[i];
        s += v * v;
    }
    __shared__ float red[8];
    for (int off = 16; off > 0; off >>= 1) s += __shfl_down(s, off, 32);
    if ((threadIdx.x & 31) == 0) red[threadIdx.x >> 5] = s;
    __syncthreads();
    if (threadIdx.x == 0) {
        float t = 0.0f;
        for (int i = 0; i < 8; i++) t += red[i];
        float inv = 1.0f / sqrtf(t);
        if (row < Nx) invz[row] = inv; else invx[row - Nx] = inv;
    }
}

// ---------------- fp32 WMMA GEMM: sim = (z xT) * invz * invx ---------------
// 128x128 tile / block of 256 threads (8 waves). Wave tile 32x64 -> 8 accums.
// K staged through LDS in 32-wide panels, padded stride 36 (conflict-free).
__global__ __launch_bounds__(256) void kgemm(const float* __restrict__ Z,
                                             const float* __restrict__ X,
                                             const float* __restrict__ invz,
                                             const float* __restrict__ invx,
                                             float* __restrict__ sim) {
    __shared__ float As[128 * 36];
    __shared__ float Bs[128 * 36];

    const int bx = blockIdx.x & 31;        // col tile
    const int by = blockIdx.x >> 5;        // row tile
    const int row0 = by * 128;
    const int col0 = bx * 128;
    const int t = threadIdx.x;
    const int w = t >> 5;
    const int lane = t & 31;
    const int rowBase = (w & 3) * 32;      // 4 waves across rows
    const int colBase = (w >> 2) * 64;     // 2 waves across cols
    const int l15 = lane & 15;
    const int lh = lane >> 4;              // K-half select (ISA 16x4 layout)

    v8f acc[2][4];
    const v8f vzero = {0.f, 0.f, 0.f, 0.f, 0.f, 0.f, 0.f, 0.f};
    for (int i = 0; i < 2; i++)
        for (int j = 0; j < 4; j++) acc[i][j] = vzero;

    for (int kc = 0; kc < Dd; kc += 32) {
        // cooperative tile load: 128 rows x 32 K as float4, coalesced
        for (int q = 0; q < 4; q++) {
            int s = t + 256 * q;
            int r = s >> 3;
            int c4 = (s & 7) * 4;
            float4 va = *(const float4*)(Z + (size_t)(row0 + r) * Dd + kc + c4);
            float4 vb = *(const float4*)(X + (size_t)(col0 + r) * Dd + kc + c4);
            *(float4*)(As + r * 36 + c4) = va;
            *(float4*)(Bs + r * 36 + c4) = vb;
        }
        __syncthreads();

        for (int kk = 0; kk < 32; kk += 4) {
            const int kidx = kk + lh * 2;   // lanes 0-15: K{0,1}; 16-31: K{2,3}
            v2f a[2], b[4];
            for (int ri = 0; ri < 2; ri++) {
                int m = rowBase + ri * 16 + l15;
                a[ri] = *(const v2f*)(As + m * 36 + kidx);
            }
            for (int ci = 0; ci < 4; ci++) {
                int n = colBase + ci * 16 + l15;
                b[ci] = *(const v2f*)(Bs + n * 36 + kidx);
            }
            for (int ri = 0; ri < 2; ri++)
                for (int ci = 0; ci < 4; ci++)
                    acc[ri][ci] = __builtin_amdgcn_wmma_f32_16x16x4_f32(
                        false, a[ri], false, b[ci], (short)0, acc[ri][ci],
                        false, false);
        }
        __syncthreads();
    }

    // epilogue: C/D layout — VGPR r: lanes0-15 M=r, lanes16-31 M=r+8, N=lane&15
    for (int ri = 0; ri < 2; ri++) {
        for (int ci = 0; ci < 4; ci++) {
            int col = col0 + colBase + ci * 16 + l15;
            float ix = invx[col];
            for (int r = 0; r < 8; r++) {
                int row = row0 + rowBase + ri * 16 + r + lh * 8;
                sim[(size_t)row * Mx + col] = acc[ri][ci][r] * invz[row] * ix;
            }
        }
    }
}

// ---------------- radix select: k-th largest (2 x 16-bit passes) -----------
__global__ void khist1(const float* __restrict__ sim, unsigned* __restrict__ h) {
    unsigned i = blockIdx.x * blockDim.x + threadIdx.x;
    unsigned stride = gridDim.x * blockDim.x;
    for (; i < TOTAL; i += stride) {
        unsigned key = f2key(sim[i]);
        atomicAdd(&h[key >> 16], 1u);
    }
}
__global__ void khist2(const float* __restrict__ sim, const int* __restrict__ sel,
                       unsigned* __restrict__ h) {
    unsigned b1 = (unsigned)sel[0];
    unsigned i = blockIdx.x * blockDim.x + threadIdx.x;
    unsigned stride = gridDim.x * blockDim.x;
    for (; i < TOTAL; i += stride) {
        unsigned key = f2key(sim[i]);
        if ((key >> 16) == b1) atomicAdd(&h[key & 0xFFFFu], 1u);
    }
}
// level 0: find top-16-bit bin + residual rank; level 1: emit dropline float
__global__ __launch_bounds__(1024) void kscan(const unsigned* __restrict__ hist,
                                              int* __restrict__ sel, int level,
                                              float* __restrict__ dropline) {
    __shared__ unsigned sc[1024];
    int t = threadIdx.x;
    unsigned target = (level == 0) ? TARGET_IDX : (unsigned)sel[1];
    unsigned s = 0;
    for (int i = 0; i < 64; i++) s += hist[t * 64 + i];
    sc[t] = s;
    __syncthreads();
    unsigned v = s;
    for (int off = 1; off < 1024; off <<= 1) {
        unsigned u = (t >= off) ? sc[t - off] : 0u;
        __syncthreads();
        v += u;
        sc[t] = v;
        __syncthreads();
    }
    unsigned excl = (t == 0) ? 0u : sc[t - 1];
    if (target >= excl && target < excl + s) {
        unsigned c = excl;
        for (int i = 0; i < 64; i++) {
            unsigned hc = hist[t * 64 + i];
            if (target < c + hc) {
                int bin = t * 64 + i;
                if (level == 0) {
                    sel[0] = bin;
                    sel[1] = (int)(target - c);
                } else {
                    unsigned key = ((unsigned)sel[0] << 16) | (unsigned)bin;
                    *dropline = key2f(key);
                }
                break;
            }
            c += hc;
        }
    }
}

// ---------------- NW forward DP: rows sequential, per-row parallel cummin --
__global__ __launch_bounds__(1024) void kdp(const float* __restrict__ sim,
                                            const float* __restrict__ dropptr,
                                            float* __restrict__ Dtab) {
    __shared__ float rowbuf[2][4100];  // padded so rowbuf[1] stays 16B-aligned
    __shared__ float wtot[32];
    __shared__ float wexc[33];
    const float drop = *dropptr;
    const int t = threadIdx.x, lane = t & 31, wid = t >> 5;
    const float BIG = 3.0e38f;

    for (int j = t; j <= Mx; j += 1024) {
        rowbuf[0][j] = 0.0f;
        Dtab[j] = 0.0f;
    }
    __syncthreads();

    int pb = 0;
    for (int i = 1; i <= Nx; i++) {
        float* prev = rowbuf[pb];
        float* cur = rowbuf[pb ^ 1];
        const float p0 = prev[0];
        const int j0 = t * 4;  // == (j-1) of this thread's first element
        float4 pr = *(const float4*)(prev + j0);
        float p4 = prev[j0 + 4];
        float4 sv = *(const float4*)(sim + (size_t)(i - 1) * Mx + j0);
        float cand0 = fminf(pr.x + (drop - sv.x), pr.y);
        float cand1 = fminf(pr.y + (drop - sv.y), pr.z);
        float cand2 = fminf(pr.z + (drop - sv.z), pr.w);
        float cand3 = fminf(pr.w + (drop - sv.w), p4);
        float l0 = cand0;
        float l1 = fminf(l0, cand1);
        float l2 = fminf(l1, cand2);
        float l3 = fminf(l2, cand3);

        // wave32 inclusive min-scan of thread minima
        float v = l3;
        for (int off = 1; off < 32; off <<= 1) {
            float u = __shfl_up(v, off, 32);
            if (lane >= off) v = fminf(v, u);
        }
        if (lane == 31) wtot[wid] = v;
        __syncthreads();
        if (wid == 0) {
            float wv = wtot[lane];
            for (int off = 1; off < 32; off <<= 1) {
                float u = __shfl_up(wv, off, 32);
                if (lane >= off) wv = fminf(wv, u);
            }
            wexc[lane + 1] = wv;
            if (lane == 0) wexc[0] = BIG;
        }
        __syncthreads();
        float prevIncl = __shfl_up(v, 1, 32);
        if (lane == 0) prevIncl = BIG;
        float base = fminf(p0, fminf(wexc[wid], prevIncl));

        float r0 = fminf(base, l0), r1 = fminf(base, l1);
        float r2 = fminf(base, l2), r3 = fminf(base, l3);
        cur[j0 + 1] = r0; cur[j0 + 2] = r1; cur[j0 + 3] = r2; cur[j0 + 4] = r3;
        size_t rb = (size_t)i * (Mx + 1);
        Dtab[rb + j0 + 1] = r0; Dtab[rb + j0 + 2] = r1;
        Dtab[rb + j0 + 3] = r2; Dtab[rb + j0 + 4] = r3;
        if (t == 0) { cur[0] = p0; Dtab[rb] = p0; }
        __syncthreads();
        pb ^= 1;
    }
}

// ---------------- traceback: sequential, L2-resident tables ----------------
__global__ void ktb(const float* __restrict__ sim, const float* __restrict__ Dtab,
                    const float* __restrict__ dropptr, float* __restrict__ out) {
    if (threadIdx.x != 0 || blockIdx.x != 0) return;
    const float drop = *dropptr;
    int i = Nx, j = Mx;
    for (int s = 0; s < Nx + Mx; s++) {
        bool origin = (i == 0) && (j == 0);
        int ci = (i > 0) ? i - 1 : 0;
        int cj = (j > 0) ? j - 1 : 0;
        float Dij = Dtab[(size_t)i * (Mx + 1) + j];
        float c = drop - sim[(size_t)ci * Mx + cj];
        bool take_diag = (i > 0) && (j > 0) &&
                         (Dij == Dtab[(size_t)ci * (Mx + 1) + cj] + c);
        bool take_up = (!take_diag) && (i > 0) &&
                       (Dij == Dtab[(size_t)ci * (Mx + 1) + j]);
        if (take_diag && !origin) out[(size_t)ci * Mx + cj] = 1.0f;
        if (!origin) {
            i -= (take_diag || take_up) ? 1 : 0;
            j -= (!take_up) ? 1 : 0;
        }
    }
}

// ---------------------------------------------------------------------------
extern "C" void kernel_launch(void* const* d_in, const int* in_sizes, int n_in,
                              void* d_out, int out_size, void* d_ws, size_t ws_size,
                              hipStream_t stream) {
    (void)in_sizes; (void)n_in; (void)out_size; (void)ws_size;
    const float* z = (const float*)d_in[2];   // text_embed  [4096,1024] f32
    const float* x = (const float*)d_in[3];   // event_embed [4096,1024] f32
    float* out = (float*)d_out;               // corresp [4096,4096] f32

    char* ws = (char*)d_ws;
    const size_t OFF_SIM  = 0;                               // 64 MB
    const size_t OFF_DTAB = (size_t)Nx * Mx * 4;             // 67.14 MB
    const size_t OFF_INVZ = OFF_DTAB + (size_t)(Nx + 1) * (Mx + 1) * 4 + 252;
    const size_t OFF_INVZ_A = OFF_INVZ & ~(size_t)255;       // 256B aligned
    const size_t OFF_INVX = OFF_INVZ_A + Nx * 4;
    const size_t OFF_H1   = OFF_INVX + Mx * 4;
    const size_t OFF_H2   = OFF_H1 + 65536 * 4;
    const size_t OFF_SEL  = OFF_H2 + 65536 * 4;
    const size_t OFF_DROP = OFF_SEL + 16;

    float*    sim   = (float*)(ws + OFF_SIM);
    float*    Dtab  = (float*)(ws + OFF_DTAB);
    float*    invz  = (float*)(ws + OFF_INVZ_A);
    float*    invx  = (float*)(ws + OFF_INVX);
    unsigned* h1    = (unsigned*)(ws + OFF_H1);
    unsigned* h2    = (unsigned*)(ws + OFF_H2);
    int*      sel   = (int*)(ws + OFF_SEL);
    float*    drop  = (float*)(ws + OFF_DROP);

    // clear select state (replays reuse ws) and the output matrix
    kzero_u<<<512, 256, 0, stream>>>((unsigned*)(ws + OFF_H1),
                                     65536u * 2 + 8u);
    kzero_f<<<4096, 256, 0, stream>>>(out, TOTAL);

    // 1) inverse row norms for both embeddings
    knorm<<<Nx + Mx, 256, 0, stream>>>(z, x, invz, invx);

    // 2) fp32 WMMA GEMM -> sim
    kgemm<<<1024, 256, 0, stream>>>(z, x, invz, invx, sim);

    // 3) radix-select dropline = k-th largest of sim
    khist1<<<2048, 256, 0, stream>>>(sim, h1);
    kscan<<<1, 1024, 0, stream>>>(h1, sel, 0, drop);
    khist2<<<2048, 256, 0, stream>>>(sim, sel, h2);
    kscan<<<1, 1024, 0, stream>>>(h2, sel, 1, drop);

    // 4) NW forward DP (sequential rows, parallel cummin per row)
    kdp<<<1, 1024, 0, stream>>>(sim, drop, Dtab);

    // 5) traceback into corresp
    ktb<<<1, 32, 0, stream>>>(sim, Dtab, drop, out);
}